// VectorQuantizer_29626684408051
// MI455X (gfx1250) — compile-verified
//
#include <hip/hip_runtime.h>
#include <hip/hip_bf16.h>
#include <math.h>

typedef __attribute__((ext_vector_type(2))) float v2f;
typedef __attribute__((ext_vector_type(8))) float v8f;

#define NUM_EMB   1024
#define EMB_DIM   64
#define HW        1024            // 32*32
#define NTOT      65536           // 64 * 32 * 32
#define QUANT_ELEMS (64*64*32*32) // 4194304

#define OUT_QUANT_OFF 1
#define OUT_PERP_OFF  (1 + QUANT_ELEMS)   // 4194305
#define OUT_IDX_OFF   (2 + QUANT_ELEMS)   // 4194306

// workspace byte offsets
#define WS_CNORM 0        // 1024 f32
#define WS_HIST  4096     // 1024 i32
#define WS_LOSS  8192     // 1 f32

// ---------------------------------------------------------------- kernel 0
__global__ __launch_bounds__(256) void vq_cnorm(const float* __restrict__ cb,
                                                float* __restrict__ cnorm) {
  int k = blockIdx.x * blockDim.x + threadIdx.x;
  if (k < NUM_EMB) {
    const float* row = cb + (size_t)k * EMB_DIM;
    float s = 0.f;
    #pragma unroll
    for (int c = 0; c < EMB_DIM; ++c) { float v = row[c]; s += v * v; }
    cnorm[k] = s;
  }
}

// ---------------------------------------------------------------- kernel 1
// Fully fused: one wave32 per tile of 16 latent rows.
//  (a) distances via V_WMMA_F32_16X16X4_F32 (fp32 exact) against all 1024
//      codes, argmin of ||c||^2 - 2 x.c per row,
//  (b) winner broadcast through LDS,
//  (c) codebook gather + quantized write + sum((q-x)^2) straight from the
//      A-matrix registers (latents are read from HBM exactly once).
__global__ __launch_bounds__(32) void vq_fused(
    const float* __restrict__ latents,   // [64, 64, 32, 32]
    const float* __restrict__ cb,        // [1024, 64]
    const float* __restrict__ cnorm,     // [1024]
    int*   __restrict__ hist,            // [1024]
    float* __restrict__ loss_accum,      // [1]
    float* __restrict__ quant_out,       // [64,64,32,32] region of d_out
    float* __restrict__ out_idx_f)       // [NTOT] (float view in d_out)
{
  const int lane = threadIdx.x;     // 0..31
  const int tile = blockIdx.x;      // 0..4095
  const int half = lane >> 4;       // 0: K={0,1}/M=v ; 1: K={2,3}/M=v+8
  const int l16  = lane & 15;

  // ---- Load A-matrix (16 rows x 64 K) once, ISA 16x4 f32 layout per chunk.
  // x_flat[n][c] = latents[b][c][h][w], n = b*1024 + hw (tiles never span b).
  const int n_row = tile * 16 + l16;
  const int b     = n_row >> 10;
  const int hw    = n_row & (HW - 1);
  const float* latbase = latents + (size_t)b * EMB_DIM * HW + hw;

  v2f a[16];
  #pragma unroll
  for (int k = 0; k < 16; ++k) {
    const int c = 4 * k + 2 * half;
    a[k].x = latbase[(size_t)c * HW];
    a[k].y = latbase[(size_t)(c + 1) * HW];
  }

  float bestv[8];
  int   besti[8];
  #pragma unroll
  for (int v = 0; v < 8; ++v) { bestv[v] = 3.4e38f; besti[v] = 0; }

  // ---- Sweep 64 tiles of 16 codes
  for (int t = 0; t < NUM_EMB / 16; ++t) {
    const int code = t * 16 + l16;                 // this lane's column N
    const float* cbrow = cb + (size_t)code * EMB_DIM + 2 * half;

    v8f acc = {};
    #pragma unroll
    for (int k = 0; k < 16; ++k) {
      // B-matrix 4x16: lane holds col N=code, K = 4k + 2*half + {0,1}
      v2f bvec = *(const v2f*)(cbrow + 4 * k);
      acc = __builtin_amdgcn_wmma_f32_16x16x4_f32(
          /*neg_a=*/false, a[k], /*neg_b=*/false, bvec,
          /*c_mod=*/(short)0, acc, /*reuse_a=*/false, /*reuse_b=*/false);
    }

    const float cn = cnorm[code];
    #pragma unroll
    for (int v = 0; v < 8; ++v) {                  // row M = v + 8*half
      const float d = cn - 2.0f * acc[v];
      if (d < bestv[v]) { bestv[v] = d; besti[v] = code; }
    }
  }

  // ---- Butterfly min-reduce across the 16 lanes sharing each row.
  // After the full xor butterfly every lane in a half holds the winner.
  __shared__ int widx[16];
  #pragma unroll
  for (int v = 0; v < 8; ++v) {
    float bv = bestv[v];
    int   bi = besti[v];
    #pragma unroll
    for (int off = 8; off >= 1; off >>= 1) {
      const float ov = __shfl_xor(bv, off, 32);
      const int   oi = __shfl_xor(bi, off, 32);
      if (ov < bv || (ov == bv && oi < bi)) { bv = ov; bi = oi; }
    }
    if (l16 == 0) {
      const int row = v + 8 * half;                // global row index in tile
      const int n   = tile * 16 + row;
      widx[row]     = bi;
      out_idx_f[n]  = (float)bi;
      atomicAdd(&hist[bi], 1);
    }
  }
  __syncthreads();

  // ---- Gather winning code for this lane's A-row; write quantized and
  //      accumulate the squared error straight from registers.
  const int kwin = widx[l16];                      // winner for row l16
  const float* qrow = cb + (size_t)kwin * EMB_DIM;
  float* qbase = quant_out + (size_t)b * EMB_DIM * HW + hw;

  float s = 0.f;
  #pragma unroll
  for (int k = 0; k < 16; ++k) {
    const int c = 4 * k + 2 * half;
    const float q0 = qrow[c];
    const float q1 = qrow[c + 1];
    qbase[(size_t)c * HW]       = q0;   // lanes 0..15 contiguous per channel
    qbase[(size_t)(c + 1) * HW] = q1;
    const float d0 = q0 - a[k].x;
    const float d1 = q1 - a[k].y;
    s += d0 * d0 + d1 * d1;
  }

  // wave-reduce loss, one atomic per tile
  #pragma unroll
  for (int off = 16; off >= 1; off >>= 1) s += __shfl_xor(s, off, 32);
  if (lane == 0) atomicAdd(loss_accum, s);
}

// ---------------------------------------------------------------- kernel 2
__global__ __launch_bounds__(1024) void vq_finalize(
    const int* __restrict__ hist,
    const float* __restrict__ loss_accum,
    float* __restrict__ out_loss,
    float* __restrict__ out_perp)
{
  __shared__ float red[1024];
  const int i = threadIdx.x;
  const float p = (float)hist[i] / (float)NTOT;
  red[i] = -p * logf(p + 1e-10f);
  __syncthreads();
  for (int off = 512; off >= 1; off >>= 1) {
    if (i < off) red[i] += red[i + off];
    __syncthreads();
  }
  if (i == 0) {
    *out_perp = expf(red[0]);
    // e_loss == q_loss numerically -> vq_loss = (1 + 0.25) * mean
    *out_loss = 1.25f * (*loss_accum) / (float)((size_t)NTOT * EMB_DIM);
  }
}

// ---------------------------------------------------------------- launch
extern "C" void kernel_launch(void* const* d_in, const int* in_sizes, int n_in,
                              void* d_out, int out_size, void* d_ws, size_t ws_size,
                              hipStream_t stream) {
  const float* latents  = (const float*)d_in[0];  // [64,64,32,32] f32
  const float* codebook = (const float*)d_in[1];  // [1024,64] f32
  float* out = (float*)d_out;
  char*  ws  = (char*)d_ws;

  float* cnorm = (float*)(ws + WS_CNORM);
  int*   hist  = (int*)  (ws + WS_HIST);
  float* loss  = (float*)(ws + WS_LOSS);

  // zero histogram + loss accumulator (graph-capturable memset node)
  hipMemsetAsync(ws + WS_HIST, 0, 4096 + 64, stream);

  vq_cnorm   <<<NUM_EMB / 256, 256, 0, stream>>>(codebook, cnorm);
  vq_fused   <<<NTOT / 16,      32, 0, stream>>>(latents, codebook, cnorm,
                                                 hist, loss,
                                                 out + OUT_QUANT_OFF,
                                                 out + OUT_IDX_OFF);
  vq_finalize<<<1,            1024, 0, stream>>>(hist, loss, out + 0,
                                                 out + OUT_PERP_OFF);
}